// _TopKSigmoid_8907762172111
// MI455X (gfx1250) — compile-verified
//
#include <hip/hip_runtime.h>
#include <math.h>

#define D_DIM 4096
#define TPB   256
#define K_SEL 64
#define NH       8            // one sub-histogram per wave
#define HSTRIDE  264          // 256 bins + 8-dword skew => +8 banks per wave (264 % 64 == 8)
#define HTOT     (NH * HSTRIDE)

// Order-preserving float -> uint key: larger float => larger key. Invertible.
__device__ __forceinline__ unsigned flip_key(unsigned u) {
  return u ^ ((unsigned)((int)u >> 31) | 0x80000000u);
}
__device__ __forceinline__ float key_to_float(unsigned k) {
  const unsigned u = (k & 0x80000000u) ? (k ^ 0x80000000u) : ~k;
  return __uint_as_float(u);
}
__device__ __forceinline__ float sigmoidf_(float v) {
  return 1.0f / (1.0f + expf(-v));
}

// Wave-0 scan: sum the 8 sub-histograms, suffix-scan 256 bins (8 bins/lane, 5 shuffle
// steps across lanes), unique crossing lane refines its 8 bins and publishes result.
__device__ __forceinline__ void scan_select(const unsigned* s_hist, unsigned* s_prefix,
                                            unsigned* s_remaining, int tid,
                                            unsigned rem, unsigned prefix) {
  if (tid < 32) {
    const unsigned b8 = (unsigned)tid * 8u;
    unsigned h[8] = {0u, 0u, 0u, 0u, 0u, 0u, 0u, 0u};
#pragma unroll
    for (int w = 0; w < NH; ++w) {
      const uint4 a = *(const uint4*)&s_hist[w * HSTRIDE + b8];
      const uint4 b = *(const uint4*)&s_hist[w * HSTRIDE + b8 + 4];
      h[0] += a.x; h[1] += a.y; h[2] += a.z; h[3] += a.w;
      h[4] += b.x; h[5] += b.y; h[6] += b.z; h[7] += b.w;
    }
    const unsigned bs = h[0] + h[1] + h[2] + h[3] + h[4] + h[5] + h[6] + h[7];
    unsigned suf = bs;                       // inclusive suffix sum across lanes
#pragma unroll
    for (int d = 1; d < 32; d <<= 1) {
      const unsigned o = __shfl_down(suf, d, 32);
      suf += (tid + d < 32) ? o : 0u;
    }
    const unsigned above = suf - bs;         // count strictly above my 8-bin block
    if (suf >= rem && above < rem) {         // unique crossing lane
      unsigned acc = above, nrem = 1u;
      int sel = 0;
      bool found = false;
#pragma unroll
      for (int b = 7; b >= 0; --b) {
        if (!found) {
          if (acc + h[b] >= rem) { sel = b; nrem = rem - acc; found = true; }
          else acc += h[b];
        }
      }
      *s_prefix = (prefix << 8) | (b8 + (unsigned)sel);
      *s_remaining = nrem;
    }
  }
}

__global__ __launch_bounds__(TPB) void topk64_sigmoid_rows(
    const float* __restrict__ x, float* __restrict__ out, int rows) {
  // Phase A: async-staged row (16 KB). Phase B (keys in VGPRs): tie-index buffer.
  __shared__ unsigned s_buf[D_DIM];
  __shared__ __align__(16) unsigned s_hist[HTOT];
  __shared__ unsigned s_prefix;
  __shared__ unsigned s_remaining;
  __shared__ unsigned s_eq_cnt;
  __shared__ unsigned s_It;

  const int row = blockIdx.x;
  if (row >= rows) return;
  const int tid = threadIdx.x;
  const unsigned whoff = ((unsigned)tid >> 5) * HSTRIDE;   // this wave's sub-histogram
  const float* __restrict__ rowp = x + (size_t)row * D_DIM;
  float* __restrict__ outp = out + (size_t)row * D_DIM;

  if (tid == 0) s_eq_cnt = 0u;

  // ---- gfx1250 async copy: 16 KB row -> LDS (ASYNCcnt), 512 B per wave-instruction ----
  {
    const unsigned lds_base = (unsigned)(unsigned long long)(size_t)&s_buf[0];
    const unsigned long long gbase = (unsigned long long)(size_t)rowp;
#pragma unroll
    for (int i = 0; i < 4; ++i) {
      const unsigned boff = (unsigned)tid * 16u + (unsigned)i * (TPB * 16u);
      const unsigned loff = lds_base + boff;
      asm volatile("global_load_async_to_lds_b128 %0, %1, %2"
                   :: "v"(loff), "v"(boff), "s"(gbase)
                   : "memory");
    }
    asm volatile("s_wait_asynccnt 0" ::: "memory");
  }
  __syncthreads();

  // ---- read row once; keys stay in VGPRs for the rest of the kernel ----
  unsigned kk[16];
#pragma unroll
  for (int i = 0; i < 4; ++i) {
    const uint4 v = ((const uint4*)s_buf)[tid + i * TPB];   // ds_load_b128
    kk[i * 4 + 0] = flip_key(v.x);
    kk[i * 4 + 1] = flip_key(v.y);
    kk[i * 4 + 2] = flip_key(v.z);
    kk[i * 4 + 3] = flip_key(v.w);
  }

  // ---- pass 0 (peeled: no prefix filter, rem/prefix are constants) ----
#pragma unroll
  for (int j = 0; j < 3; ++j) {                     // zero 8 sub-histograms (b128)
    const int i4 = tid + j * TPB;
    if (i4 < HTOT / 4) ((uint4*)s_hist)[i4] = make_uint4(0u, 0u, 0u, 0u);
  }
  __syncthreads();                                  // also fences the s_buf reads above
#pragma unroll
  for (int e = 0; e < 16; ++e)
    atomicAdd(&s_hist[whoff + (kk[e] >> 24)], 1u);
  __syncthreads();
  scan_select(s_hist, &s_prefix, &s_remaining, tid, K_SEL, 0u);
  __syncthreads();

  // ---- passes 1..3: prefix-filtered radix from registers ----
#pragma unroll 1
  for (int pass = 1; pass < 4; ++pass) {
#pragma unroll
    for (int j = 0; j < 3; ++j) {
      const int i4 = tid + j * TPB;
      if (i4 < HTOT / 4) ((uint4*)s_hist)[i4] = make_uint4(0u, 0u, 0u, 0u);
    }
    __syncthreads();
    const unsigned prefix = s_prefix;
    const unsigned rem = s_remaining;
    const int shift = 24 - 8 * pass;
#pragma unroll
    for (int e = 0; e < 16; ++e) {
      const unsigned k = kk[e];
      if ((k >> (shift + 8)) == prefix)
        atomicAdd(&s_hist[whoff + ((k >> shift) & 0xFFu)], 1u);
    }
    __syncthreads();
    scan_select(s_hist, &s_prefix, &s_remaining, tid, rem, prefix);
    __syncthreads();
  }

  const unsigned Kt = s_prefix;      // exact 32-bit key of the k-th largest
  const unsigned remT = s_remaining; // #elements equal to Kt to keep (lowest indices win)

  // ---- collect tie indices (reusing s_buf; capacity D_DIM => always correct) ----
#pragma unroll
  for (int i = 0; i < 4; ++i) {
    const int base = tid * 4 + i * (TPB * 4);
#pragma unroll
    for (int c = 0; c < 4; ++c) {
      if (kk[i * 4 + c] == Kt) {
        const unsigned p = atomicAdd(&s_eq_cnt, 1u);
        s_buf[p] = (unsigned)(base + c);
      }
    }
  }
  __syncthreads();
  if (tid == 0) {
    const unsigned n = s_eq_cnt;     // >= remT >= 1; typically 1
    unsigned It = 0u;
    for (unsigned t = 0; t < remT; ++t) {   // remT-th smallest tie index
      unsigned mv = 0xFFFFFFFFu, mi = 0u;
      for (unsigned j = 0; j < n; ++j) {
        const unsigned v = s_buf[j];
        if (v < mv) { mv = v; mi = j; }
      }
      s_buf[mi] = 0xFFFFFFFFu;
      It = mv;
    }
    s_It = It;
  }
  __syncthreads();
  const unsigned It = s_It;

  // ---- write full output row: sigmoid(top-64), zero elsewhere (b128 stores) ----
#pragma unroll
  for (int i = 0; i < 4; ++i) {
    const int base = tid * 4 + i * (TPB * 4);
    float4 o;
    float* op = (float*)&o;
#pragma unroll
    for (int c = 0; c < 4; ++c) {
      const unsigned k = kk[i * 4 + c];
      const unsigned col = (unsigned)(base + c);
      float val = 0.0f;
      if (k > Kt || (k == Kt && col <= It))   // exact top_k membership, O(1)
        val = sigmoidf_(key_to_float(k));
      op[c] = val;
    }
    *(float4*)&outp[base] = o;
  }
}

extern "C" void kernel_launch(void* const* d_in, const int* in_sizes, int n_in,
                              void* d_out, int out_size, void* d_ws, size_t ws_size,
                              hipStream_t stream) {
  (void)n_in; (void)d_ws; (void)ws_size; (void)out_size;
  const float* x = (const float*)d_in[0];
  float* out = (float*)d_out;
  const int rows = in_sizes[0] / D_DIM;   // 16384
  topk64_sigmoid_rows<<<dim3(rows), dim3(TPB), 0, stream>>>(x, out, rows);
}